// Pfaffian_48584670052803
// MI455X (gfx1250) — compile-verified
//
#include <hip/hip_runtime.h>
#include <math.h>
#include <stdint.h>

// Problem constants (match reference)
#define NORB 128
#define NA   32
#define NB   32
#define NDIM 64            // Pfaffian matrix is 64x64 complex128
#define MSTR 65            // padded LDS row stride (double2 units)
#define NTHREADS 256       // 8 wave32 per workgroup

#define PF_PI 3.14159265358979323846

// ---- CDNA5 async global->LDS path (ASYNCcnt-tracked) -----------------------
// Toolchain's builtin signature (from diagnostics):
//   void __builtin_amdgcn_global_load_async_to_lds_b128(
//       __attribute__((vector_size(16))) int __AS1* src,
//       __attribute__((vector_size(16))) int __AS3* dst, int imm_off, int cpol)
typedef int pf_v4i __attribute__((vector_size(16)));
typedef __attribute__((address_space(1))) pf_v4i* gptr_v4i;
typedef __attribute__((address_space(3))) pf_v4i* lptr_v4i;

__device__ __forceinline__ void async_gather16(const void* gsrc, void* ldst) {
#if defined(__has_builtin) && __has_builtin(__builtin_amdgcn_global_load_async_to_lds_b128)
  __builtin_amdgcn_global_load_async_to_lds_b128(
      (gptr_v4i)(uintptr_t)gsrc,
      (lptr_v4i)(uint32_t)(uintptr_t)ldst,
      0, 0);
#else
  // Generic->LDS: low 32 bits of a flat LDS pointer are the wave-relative
  // LDS byte offset (ISA 10.2 aperture mapping).
  uint32_t loff = (uint32_t)(uintptr_t)ldst;
  uint64_t ga   = (uint64_t)(uintptr_t)gsrc;
  asm volatile("global_load_async_to_lds_b128 %0, %1, off"
               :: "v"(loff), "v"(ga) : "memory");
#endif
}

__device__ __forceinline__ void wait_async0() {
#if defined(__has_builtin) && __has_builtin(__builtin_amdgcn_s_wait_asynccnt)
  __builtin_amdgcn_s_wait_asynccnt(0);
#else
  asm volatile("s_wait_asynccnt 0" ::: "memory");
#endif
}

// ---------------------------------------------------------------------------
// One workgroup per batch sample. Matrix resident in LDS; sequential
// Parlett-Reid rank-2 reduction with greedy pivoting, complex128 arithmetic.
__global__ void __launch_bounds__(NTHREADS)
pfaffian_kernel(const int*    __restrict__ s,     // (B, 256) int32
                const double* __restrict__ Waa,   // (128,128) complex -> 2 doubles/elt
                const double* __restrict__ Wbb,
                const double* __restrict__ Phi,
                double*       __restrict__ out)   // (B,) complex -> interleaved doubles
{
  extern __shared__ double2 Mdyn[];               // NDIM * MSTR double2 (padded)
  __shared__ int    ra[NA], rb[NB];
  __shared__ double redmag[NDIM];
  __shared__ int    s_jmax;
  __shared__ int    s_sing;
  __shared__ double s_px, s_py, s_den;

  const int tid = threadIdx.x;
  const int b   = blockIdx.x;
  const int* srow = s + (size_t)b * (2 * NORB);

  // Warm shared operand matrices into L2 (emits global_prefetch_b8).
  if (tid < 64) {
    __builtin_prefetch(Waa + (size_t)tid * 512, 0, 1);
    __builtin_prefetch(Wbb + (size_t)tid * 512, 0, 1);
    __builtin_prefetch(Phi + (size_t)tid * 512, 0, 1);
  }

  // Occupied-orbital index lists (32 set bits each, in order).
  if (tid == 0) {
    int c = 0;
    for (int o = 0; o < NORB; ++o) if (srow[o] != 0) ra[c++] = o;
    s_sing = 0;
  }
  if (tid == 32) {
    int c = 0;
    for (int o = 0; o < NORB; ++o) if (srow[NORB + o] != 0) rb[c++] = o;
  }
  __syncthreads();

  // ---- Build M --------------------------------------------------------------
  // Upper-right block A12 = Phi[ra,:][:,rb]: pure 16B gather -> async to LDS.
  for (int idx = tid; idx < NA * NB; idx += NTHREADS) {
    const int i = idx >> 5, j = idx & 31;
    const double* g = Phi + 2 * ((size_t)ra[i] * NORB + (size_t)rb[j]);
    async_gather16(g, &Mdyn[i * MSTR + (NA + j)]);
  }
  // A11 = 0.5*(Waa[ri,rj] - Waa[rj,ri])  (overlaps with in-flight async gather)
  for (int idx = tid; idx < NA * NA; idx += NTHREADS) {
    const int i = idx >> 5, j = idx & 31;
    const size_t p = 2 * ((size_t)ra[i] * NORB + (size_t)ra[j]);
    const size_t q = 2 * ((size_t)ra[j] * NORB + (size_t)ra[i]);
    Mdyn[i * MSTR + j] =
        make_double2(0.5 * (Waa[p] - Waa[q]), 0.5 * (Waa[p + 1] - Waa[q + 1]));
  }
  // A22 = 0.5*(Wbb[ri,rj] - Wbb[rj,ri])
  for (int idx = tid; idx < NB * NB; idx += NTHREADS) {
    const int i = idx >> 5, j = idx & 31;
    const size_t p = 2 * ((size_t)rb[i] * NORB + (size_t)rb[j]);
    const size_t q = 2 * ((size_t)rb[j] * NORB + (size_t)rb[i]);
    Mdyn[(NA + i) * MSTR + (NA + j)] =
        make_double2(0.5 * (Wbb[p] - Wbb[q]), 0.5 * (Wbb[p + 1] - Wbb[q + 1]));
  }
  wait_async0();
  __syncthreads();
  // Lower-left block = -A12^T (reads the async-gathered block from LDS).
  for (int idx = tid; idx < NA * NB; idx += NTHREADS) {
    const int i = idx >> 5, j = idx & 31;          // element (NA+i, j)
    const double2 t = Mdyn[j * MSTR + (NA + i)];
    Mdyn[(NA + i) * MSTR + j] = make_double2(-t.x, -t.y);
  }
  __syncthreads();

  // ---- Parlett-Reid reduction with pivoting --------------------------------
  double lr = 0.0, li = 0.0;                       // log accumulator (thread 0)
  for (int it = 0; it < NDIM / 2; ++it) {
    const int k = 2 * it;

    // 1) |M[k,j]|^2 for j>k (monotone in |.| -> same argmax & first-tie)
    if (tid < NDIM) {
      double m = -1.0;
      if (tid > k) {
        const double2 t = Mdyn[k * MSTR + tid];
        m = t.x * t.x + t.y * t.y;
      }
      redmag[tid] = m;
    }
    __syncthreads();

    // 2) first-maximum argmax (matches jnp.argmax) by thread 0
    if (tid == 0) {
      int jm = 0; double best = redmag[0];
      for (int j = 1; j < NDIM; ++j) {
        const double v = redmag[j];
        if (v > best) { best = v; jm = j; }
      }
      s_jmax = jm;
      if (jm != k + 1) li += PF_PI;                // log(-1+0j) = i*pi per swap
    }
    __syncthreads();

    const int jm = s_jmax;                         // uniform across block
    if (jm != k + 1) {
      if (tid < NDIM) {                            // swap rows k+1 <-> jm
        const double2 a = Mdyn[(k + 1) * MSTR + tid];
        const double2 c = Mdyn[jm * MSTR + tid];
        Mdyn[(k + 1) * MSTR + tid] = c;
        Mdyn[jm * MSTR + tid] = a;
      }
      __syncthreads();
      if (tid < NDIM) {                            // swap cols k+1 <-> jm
        const double2 a = Mdyn[tid * MSTR + (k + 1)];
        const double2 c = Mdyn[tid * MSTR + jm];
        Mdyn[tid * MSTR + (k + 1)] = c;
        Mdyn[tid * MSTR + jm] = a;
      }
      __syncthreads();
    }

    // 3) pivot and complex log accumulation
    if (tid == 0) {
      const double2 p = Mdyn[k * MSTR + (k + 1)];
      const double den = p.x * p.x + p.y * p.y;
      if (den < 1e-32) {                           // |pivot| < 1e-16
        s_sing = 1;
      } else {
        lr += 0.5 * log(den);                      // log|p|
        li += atan2(p.y, p.x);                     // arg(p), principal branch
        s_px = p.x; s_py = p.y; s_den = den;
      }
    }
    __syncthreads();
    if (s_sing) break;                             // uniform exit

    const double px = s_px, py = s_py, den = s_den;

    // 4) rank-2 update on trailing block: M[i,j] -= (u_i v_j - v_i u_j)/p
    //    (rows k,k+1 are read-only here -> no copies needed)
    for (int idx = tid; idx < NDIM * NDIM; idx += NTHREADS) {
      const int i = idx >> 6, j = idx & 63;
      if (i >= k + 2 && j >= k + 2) {
        const double2 ui = Mdyn[k * MSTR + i];
        const double2 uj = Mdyn[k * MSTR + j];
        const double2 vi = Mdyn[(k + 1) * MSTR + i];
        const double2 vj = Mdyn[(k + 1) * MSTR + j];
        const double wx = (ui.x * vj.x - ui.y * vj.y) - (vi.x * uj.x - vi.y * uj.y);
        const double wy = (ui.x * vj.y + ui.y * vj.x) - (vi.x * uj.y + vi.y * uj.x);
        const double dx = (wx * px + wy * py) / den;
        const double dy = (wy * px - wx * py) / den;
        const double2 m = Mdyn[i * MSTR + j];
        Mdyn[i * MSTR + j] = make_double2(m.x - dx, m.y - dy);
      }
    }
    __syncthreads();

    // 5) re-antisymmetrize (each (i<j) pair owned by exactly one lane)
    for (int idx = tid; idx < NDIM * NDIM; idx += NTHREADS) {
      const int i = idx >> 6, j = idx & 63;
      if (i < j) {
        const double2 a = Mdyn[i * MSTR + j];
        const double2 c = Mdyn[j * MSTR + i];
        const double nx = 0.5 * (a.x - c.x), ny = 0.5 * (a.y - c.y);
        Mdyn[i * MSTR + j] = make_double2(nx, ny);
        Mdyn[j * MSTR + i] = make_double2(-nx, -ny);
      } else if (i == j) {
        Mdyn[i * MSTR + j] = make_double2(0.0, 0.0);
      }
    }
    __syncthreads();
  }

  if (tid == 0) {
    if (s_sing) { out[2 * b] = -INFINITY; out[2 * b + 1] = 0.0; }
    else        { out[2 * b] = lr;        out[2 * b + 1] = li;  }
  }
}

// ---------------------------------------------------------------------------
extern "C" void kernel_launch(void* const* d_in, const int* in_sizes, int n_in,
                              void* d_out, int out_size, void* d_ws, size_t ws_size,
                              hipStream_t stream) {
  (void)n_in; (void)out_size; (void)d_ws; (void)ws_size;
  const int*    s   = (const int*)d_in[0];
  const double* Waa = (const double*)d_in[1];   // complex128 as double pairs
  const double* Wbb = (const double*)d_in[2];
  const double* Phi = (const double*)d_in[3];
  double*       out = (double*)d_out;           // complex128 interleaved

  const int nb = in_sizes[0] / (2 * NORB);      // BATCH = 1024
  const size_t shbytes = (size_t)NDIM * MSTR * sizeof(double2); // ~66.5 KB

  pfaffian_kernel<<<dim3(nb), dim3(NTHREADS), shbytes, stream>>>(
      s, Waa, Wbb, Phi, out);
}